// SCINet_Tree_59639915872722
// MI455X (gfx1250) — compile-verified
//
#include <hip/hip_runtime.h>

// SCINet tree at layer 3 over a length-4096 first axis is the identity
// permutation (even/odd split + zip is self-inverse when every level has an
// even length: 4096 -> 2048 -> 1024 -> 512). The kernel is therefore a pure
// 256 MB streaming copy: 0 FLOPs, 512 MB of HBM traffic, ~22 us floor at
// 23.3 TB/s. Optimal path on MI455X: wave32-coalesced b128 loads/stores with
// non-temporal hints (data set >> 192 MB L2), plus a demonstration slice on
// the CDNA5 async Global<->LDS data mover (ASYNCcnt) path.

typedef __attribute__((ext_vector_type(4))) float f32x4;
typedef __attribute__((ext_vector_type(4))) int   i32x4;

#if defined(__HIP_DEVICE_COMPILE__) &&                                  \
    __has_builtin(__builtin_amdgcn_global_load_async_to_lds_b128) &&    \
    __has_builtin(__builtin_amdgcn_global_store_async_from_lds_b128) && \
    __has_builtin(__builtin_amdgcn_s_wait_asynccnt)
#define SCI_HAS_ASYNC_LDS 1
// Pointer flavors the async-LDS builtins expect (from hipcc diagnostics:
// parameter type is a 4 x i32 vector pointer; AS1 = global, AS3 = LDS).
typedef __attribute__((address_space(1))) i32x4 g_i32x4;
typedef __attribute__((address_space(3))) i32x4 s_i32x4;
#else
#define SCI_HAS_ASYNC_LDS 0
#endif

__global__ __launch_bounds__(256) void scinet_identity_copy(
    const f32x4* __restrict__ src, f32x4* __restrict__ dst, long long n4) {
  const long long tid    = (long long)blockIdx.x * blockDim.x + threadIdx.x;
  const long long stride = (long long)gridDim.x * blockDim.x;

#if SCI_HAS_ASYNC_LDS
  // CDNA5 async data-mover route for 1/64 of the blocks: 16B/lane per
  // instruction, LDS staging, tracked with ASYNCcnt. Same index mapping as
  // the direct path, so coverage is identical and the result is exact.
  __shared__ f32x4 stage[256];
  if ((blockIdx.x & 63) == 0) {
    for (long long i = tid; i < n4; i += stride) {
      __builtin_amdgcn_global_load_async_to_lds_b128(
          (g_i32x4*)(src + i),
          (s_i32x4*)&stage[threadIdx.x],
          /*offset=*/0, /*cpol=*/0);
      __builtin_amdgcn_s_wait_asynccnt(0);   // per-lane LDS slot now valid
      __builtin_amdgcn_global_store_async_from_lds_b128(
          (g_i32x4*)(dst + i),
          (s_i32x4*)&stage[threadIdx.x],
          /*offset=*/0, /*cpol=*/0);
      __builtin_amdgcn_s_wait_asynccnt(0);   // LDS slot reusable next iter
    }
    return;
  }
#endif

  // Main path: 4x-unrolled non-temporal b128 streaming copy. Four loads in
  // flight per thread before the stores -> enough MLP to saturate HBM.
  long long i = tid;
  for (; i + 3 * stride < n4; i += 4 * stride) {
    f32x4 v0 = __builtin_nontemporal_load(src + i);
    f32x4 v1 = __builtin_nontemporal_load(src + i + stride);
    f32x4 v2 = __builtin_nontemporal_load(src + i + 2 * stride);
    f32x4 v3 = __builtin_nontemporal_load(src + i + 3 * stride);
    __builtin_nontemporal_store(v0, dst + i);
    __builtin_nontemporal_store(v1, dst + i + stride);
    __builtin_nontemporal_store(v2, dst + i + 2 * stride);
    __builtin_nontemporal_store(v3, dst + i + 3 * stride);
  }
  for (; i < n4; i += stride) {
    __builtin_nontemporal_store(__builtin_nontemporal_load(src + i), dst + i);
  }
}

// Scalar tail for element counts not divisible by 4 (not hit for this shape:
// 4096*32*512 is a multiple of 4, but kept for robustness).
__global__ void scinet_tail_copy(const float* __restrict__ src,
                                 float* __restrict__ dst,
                                 long long start, long long n) {
  long long i = start + (long long)blockIdx.x * blockDim.x + threadIdx.x;
  if (i < n) dst[i] = src[i];
}

extern "C" void kernel_launch(void* const* d_in, const int* in_sizes, int n_in,
                              void* d_out, int out_size, void* d_ws, size_t ws_size,
                              hipStream_t stream) {
  (void)n_in; (void)d_ws; (void)ws_size; (void)out_size;

  const float* x   = (const float*)d_in[0];
  float*       out = (float*)d_out;

  const long long n  = (long long)in_sizes[0];   // 4096*32*512 = 67,108,864
  const long long n4 = n >> 2;                   // 16,777,216 float4s

  const int block = 256;                         // 8 wave32 waves / block
  long long want  = (n4 + block - 1) / block;
  const int grid  = (int)(want < 4096 ? (want > 0 ? want : 1) : 4096);

  scinet_identity_copy<<<grid, block, 0, stream>>>(
      (const f32x4*)x, (f32x4*)out, n4);

  const long long tail_start = n4 << 2;
  const long long tail = n - tail_start;
  if (tail > 0) {
    const int tgrid = (int)((tail + block - 1) / block);
    scinet_tail_copy<<<tgrid, block, 0, stream>>>(x, out, tail_start, n);
  }
}